// TemplateAttacker_66597762892256
// MI455X (gfx1250) — compile-verified
//
#include <hip/hip_runtime.h>
#include <math.h>

#define B_SZ 8192
#define T_SZ 1400
#define C_SZ 256
#define S_SZ 64
#define BN   128          // B-tile per block in maha_kernel
#define DPAD (BN + 2)     // LDS row pad for D tile
#define MPAD 66           // LDS row pad for M tile (even -> 8B-aligned v2f loads)

typedef __attribute__((ext_vector_type(2))) float v2f;
typedef __attribute__((ext_vector_type(8))) float v8f;

// ---------------------------------------------------------------------------
// Kernel 1: gather x at the 64 points of interest, stored transposed [S][B]
// so the GEMM kernel's LDS fills are coalesced.
// ---------------------------------------------------------------------------
__global__ void gather_kernel(const float* __restrict__ x,
                              const int* __restrict__ idx,
                              float* __restrict__ xg) {
    int s = blockIdx.x;                                   // 0..63
    int b = blockIdx.y * blockDim.x + threadIdx.x;        // 0..8191
    int t = idx[s];
    xg[(size_t)s * B_SZ + b] = x[(size_t)b * T_SZ + t];
}

// ---------------------------------------------------------------------------
// Kernel 2: per class c, invert lower-triangular L_c by forward substitution
// (thread j owns column j), and compute const_c = -0.5*S*log(2*pi) - sum log Lii.
// 64 threads (2 waves) per block, one block per class.
// ---------------------------------------------------------------------------
__global__ void prep_kernel(const float* __restrict__ chol,
                            float* __restrict__ minv,
                            float* __restrict__ cconst) {
    __shared__ float Ls[S_SZ * 65];   // padded: stride 65 -> conflict-free
    __shared__ float Mc[S_SZ * 65];   // column j of L^-1 lives at Mc[j*65 + i]
    __shared__ float red[S_SZ];

    int c   = blockIdx.x;
    int tid = threadIdx.x;            // 0..63
    const float* L = chol + (size_t)c * S_SZ * S_SZ;

    for (int i = 0; i < S_SZ; ++i)
        Ls[i * 65 + tid] = L[i * S_SZ + tid];
    __syncthreads();

    int j = tid;
    for (int i = 0; i < j; ++i) Mc[j * 65 + i] = 0.0f;
    Mc[j * 65 + j] = 1.0f / Ls[j * 65 + j];
    for (int i = j + 1; i < S_SZ; ++i) {
        float s = 0.0f;
        for (int k = j; k < i; ++k) s += Ls[i * 65 + k] * Mc[j * 65 + k];
        Mc[j * 65 + i] = -s / Ls[i * 65 + i];
    }
    red[j] = logf(Ls[j * 65 + j]);
    __syncthreads();

    float* mout = minv + (size_t)c * S_SZ * S_SZ;         // row-major M_c
    for (int it = 0; it < S_SZ; ++it) {
        int e  = it * S_SZ + tid;                          // e = i*64 + jj
        int i  = e >> 6;
        int jj = e & 63;
        mout[e] = Mc[jj * 65 + i];
    }
    if (tid == 0) {
        float hl = 0.0f;
        for (int k = 0; k < S_SZ; ++k) hl += red[k];
        const float LOG2PI = 1.8378770664093453f;
        cconst[c] = -0.5f * (float)S_SZ * LOG2PI - hl;
    }
}

// ---------------------------------------------------------------------------
// Kernel 3: per (class c, B-tile of 128): Z = M_c * (X_g - mu_c) via fp32 WMMA
// (V_WMMA_F32_16X16X4_F32), then maha[b] = sum_m Z[m][b]^2 reduced with
// ds_add_f32 LDS atomics, then out[b*C + c] = -0.5*maha + const_c.
// 256 threads = 8 wave32; wave w owns M-strip (w&3)*16, N-half (w>>2)*64.
// ---------------------------------------------------------------------------
__global__ void __launch_bounds__(256)
maha_kernel(const float* __restrict__ xg,
            const float* __restrict__ means,
            const float* __restrict__ minv,
            const float* __restrict__ cconst,
            float* __restrict__ out) {
    __shared__ float Ms[S_SZ * MPAD];     // M_c, padded rows
    __shared__ float Ds[S_SZ * DPAD];     // D = X_g - mu, [s][b]
    __shared__ float maha_s[BN];

    int c   = blockIdx.y;
    int b0  = blockIdx.x * BN;
    int tid = threadIdx.x;

    if (tid < BN) maha_s[tid] = 0.0f;

    // Load M_c (64x64) into padded LDS, coalesced.
    const float* M = minv + (size_t)c * S_SZ * S_SZ;
    #pragma unroll
    for (int it = 0; it < (S_SZ * S_SZ) / 256; ++it) {    // 16 iters
        int e = it * 256 + tid;
        Ms[(e >> 6) * MPAD + (e & 63)] = M[e];
    }
    // Load D tile (64 x 128) with mean subtraction, coalesced per row.
    const float* mu = means + (size_t)c * S_SZ;
    #pragma unroll
    for (int it = 0; it < (S_SZ * BN) / 256; ++it) {      // 32 iters
        int e = it * 256 + tid;
        int s = e >> 7;
        int jcol = e & (BN - 1);
        Ds[s * DPAD + jcol] = xg[(size_t)s * B_SZ + b0 + jcol] - mu[s];
    }
    __syncthreads();

    int wave = tid >> 5;
    int lane = tid & 31;
    int mstrip = (wave & 3) * 16;                  // 0,16,32,48
    int nhalf  = (wave >> 2) * 64;                 // 0 or 64

    // A fragments: ISA 32-bit A 16x4 layout — lanes 0-15: K=0,1; lanes 16-31: K=2,3.
    int arow = mstrip + (lane & 15);
    int kadd = (lane >> 4) * 2;                    // 0 or 2
    v2f afrag[16];
    #pragma unroll
    for (int kb = 0; kb < 16; ++kb) {
        const float* p = &Ms[arow * MPAD + kb * 4 + kadd]; // even index -> 8B aligned
        afrag[kb] = *(const v2f*)p;
    }

    int ncol = nhalf + (lane & 15);
    #pragma unroll
    for (int nt = 0; nt < 4; ++nt) {
        int col = ncol + nt * 16;
        v8f acc = {};
        #pragma unroll
        for (int kb = 0; kb < 16; ++kb) {
            int k0 = kb * 4 + kadd;                // B rows: lanes<16 -> K0,K1; lanes>=16 -> K2,K3
            v2f bfrag;
            bfrag.x = Ds[(k0 + 0) * DPAD + col];
            bfrag.y = Ds[(k0 + 1) * DPAD + col];
            acc = __builtin_amdgcn_wmma_f32_16x16x4_f32(
                /*neg_a=*/false, afrag[kb], /*neg_b=*/false, bfrag,
                /*c_mod=*/(short)0, acc, /*reuse_a=*/false, /*reuse_b=*/false);
        }
        // Per-lane partial sum of squares over this fragment's 8 M-rows.
        float ss = 0.0f;
        #pragma unroll
        for (int r = 0; r < 8; ++r) ss += acc[r] * acc[r];
        atomicAdd(&maha_s[col], ss);               // ds_add_f32
    }
    __syncthreads();

    if (tid < BN) {
        float lp = -0.5f * maha_s[tid] + cconst[c];
        out[(size_t)(b0 + tid) * C_SZ + c] = lp;   // output shape (B, C)
    }
}

// ---------------------------------------------------------------------------
extern "C" void kernel_launch(void* const* d_in, const int* in_sizes, int n_in,
                              void* d_out, int out_size, void* d_ws, size_t ws_size,
                              hipStream_t stream) {
    const float* x     = (const float*)d_in[0];   // (B,1,T)
    const int*   idx   = (const int*)d_in[1];     // (S,)
    const float* means = (const float*)d_in[2];   // (C,S)
    const float* chol  = (const float*)d_in[3];   // (C,S,S)
    float*       out   = (float*)d_out;           // (B,C)

    char* ws = (char*)d_ws;
    size_t minv_bytes = (size_t)C_SZ * S_SZ * S_SZ * sizeof(float);  // 4 MB
    float* minv   = (float*)ws;
    float* cconst = (float*)(ws + minv_bytes);                        // 1 KB
    float* xgath  = (float*)(ws + minv_bytes + 4096);                 // 2 MB

    gather_kernel<<<dim3(S_SZ, B_SZ / 256), 256, 0, stream>>>(x, idx, xgath);
    prep_kernel<<<dim3(C_SZ), S_SZ, 0, stream>>>(chol, minv, cconst);
    maha_kernel<<<dim3(B_SZ / BN, C_SZ), 256, 0, stream>>>(xgath, means, minv, cconst, out);
}